// Pendulum2_L_dae_86784109183229
// MI455X (gfx1250) — compile-verified
//
#include <hip/hip_runtime.h>

// Pendulum2 constrained-dynamics (DAE) forward, closed form per row.
// coords row: [x0,y0,x1,y1, vx0,vy0,vx1,vy1]  ->  out row: [v(4), a(4)]
// m = [1, 50], l = [1, 1], g = 10.
//
// Memory-bound streaming kernel: 32B in + 32B out per row, ~60 FLOPs.
// One row per lane, 128-bit vector loads/stores, default (RT) cache hints
// since the 128MB working set fits in the 192MB L2 across graph replays.

__global__ __launch_bounds__(256) void pend2_dae_kernel(
    const float* __restrict__ coords, float* __restrict__ out, int n)
{
    int i = blockIdx.x * blockDim.x + threadIdx.x;
    if (i >= n) return;

    const float4* __restrict__ in4 = reinterpret_cast<const float4*>(coords);
    float4 X = in4[2 * i + 0];   // x0 y0 x1 y1
    float4 V = in4[2 * i + 1];   // vx0 vy0 vx1 vy1

    const float minv2 = 1.0f / 50.0f;   // M^-1 for body 1 (m1 = 50)
    const float F1 = -10.0f;            // -m0*g
    const float F3 = -500.0f;           // -m1*g

    float x0 = X.x, y0 = X.y, x1 = X.z, y1 = X.w;
    float vx0 = V.x, vy0 = V.y, vx1 = V.z, vy1 = V.w;

    float dx  = x0 - x1,  dy  = y0 - y1;
    float dvx = vx0 - vx1, dvy = vy0 - vy1;

    // phi_q row1 = [2x0, 2y0, 0, 0]; row2 = [2dx, 2dy, -2dx, -2dy]
    float p1x = 2.0f * x0, p1y = 2.0f * y0;
    float p2a = 2.0f * dx, p2b = 2.0f * dy;

    // phiq_Minv row2 tail entries: -2dx/50, -2dy/50
    float q2c = -p2a * minv2, q2d = -p2b * minv2;

    // L = phiq_Minv @ phi_q^T (2x2, symmetric since M^-1 is diagonal)
    float L11 = p1x * p1x + p1y * p1y;
    float L12 = p1x * p2a + p1y * p2b;
    float L22 = (p2a * p2a + p2b * p2b) + (-q2c * p2a) + (-q2d * p2b);

    // R = phiq_Minv @ F + phi_qq @ v
    // (keep the F1/F3 gravity terms in R2 explicit to track reference rounding)
    float R1 = p1y * F1 + 2.0f * (vx0 * vx0 + vy0 * vy0);
    float R2 = p2b * F1 + q2d * F3 + 2.0f * (dvx * dvx + dvy * dvy);

    // lam = inv(L) @ R  (2x2 adjugate / det, IEEE f32 division)
    float det    = L11 * L22 - L12 * L12;
    float invdet = 1.0f / det;
    float lam1 = (L22 * R1 - L12 * R2) * invdet;
    float lam2 = (L11 * R2 - L12 * R1) * invdet;

    // a = M^-1 * (F - phi_q^T @ lam)
    float ax0 = -(p1x * lam1 + p2a * lam2);
    float ay0 = F1 - (p1y * lam1 + p2b * lam2);
    float ax1 = minv2 * (p2a * lam2);
    float ay1 = minv2 * (F3 + p2b * lam2);

    float4* __restrict__ o4 = reinterpret_cast<float4*>(out);
    o4[2 * i + 0] = V;
    o4[2 * i + 1] = make_float4(ax0, ay0, ax1, ay1);
}

extern "C" void kernel_launch(void* const* d_in, const int* in_sizes, int n_in,
                              void* d_out, int out_size, void* d_ws, size_t ws_size,
                              hipStream_t stream) {
    (void)n_in; (void)out_size; (void)d_ws; (void)ws_size;
    // d_in[0] = t (unused, autonomous system); d_in[1] = coords [bs, 8] f32
    const float* coords = (const float*)d_in[1];
    float* out = (float*)d_out;
    int n = in_sizes[1] / 8;   // 2,097,152 rows

    const int block = 256;                     // 8 wave32 waves per block
    const int grid  = (n + block - 1) / block; // 8192 blocks
    pend2_dae_kernel<<<grid, block, 0, stream>>>(coords, out, n);
}